// JensenShannonDivergence_18305150615593
// MI455X (gfx1250) — compile-verified
//
#include <hip/hip_runtime.h>
#include <stdint.h>

// ---------------- tiling parameters ----------------
#define TILE   64                 // 64x64 output tile per block
#define DTOT   512
#define DC     128                // D-chunk staged per buffer
#define NCHUNK (DTOT / DC)        // 4
#define STR    132                // padded LDS row stride (floats): kills bank conflicts, 16B-aligned rows
#define ROWF   (TILE * STR)       // floats per matrix per buffer  (8448)
#define BUFF   (2 * ROWF)         // floats per double-buffer slot (a+b)
#define RENT_OFF (2 * BUFF)       // row-entropy scratch offset (floats)
#define SMEM_FLOATS (RENT_OFF + 128)
#define SMEM_BYTES  (SMEM_FLOATS * 4)   // 135,680 B  (< 320KB WGP LDS)

typedef float v2f __attribute__((ext_vector_type(2)));
typedef float v8f __attribute__((ext_vector_type(8)));
typedef int   v4i __attribute__((ext_vector_type(4)));

#define AS1 __attribute__((address_space(1)))
#define AS3 __attribute__((address_space(3)))

#if __has_builtin(__builtin_amdgcn_global_load_async_to_lds_b128)
#define HAVE_ASYNC 1
#endif
#if __has_builtin(__builtin_amdgcn_s_wait_asynccnt)
#define WAIT_ASYNC(n) __builtin_amdgcn_s_wait_asynccnt(n)
#else
#define WAIT_ASYNC(n) asm volatile("s_wait_asynccnt %0" ::"i"(n) : "memory")
#endif
#if __has_builtin(__builtin_amdgcn_wmma_f32_16x16x4_f32)
#define HAVE_WMMA4 1
#endif

__device__ __forceinline__ void async_b128(const void* g, void* l) {
#ifdef HAVE_ASYNC
  // signature: (v4i addrspace(1)* gsrc, v4i addrspace(3)* ldst, imm offset, imm cpol)
  __builtin_amdgcn_global_load_async_to_lds_b128((AS1 v4i*)g, (AS3 v4i*)l, 0, 0);
#else
  unsigned loff = (unsigned)(uintptr_t)(AS3 void*)l;
  unsigned long long ga = (unsigned long long)(uintptr_t)g;
  asm volatile("global_load_async_to_lds_b128 %0, %1, off" ::"v"(loff), "v"(ga) : "memory");
#endif
}

__device__ __forceinline__ float flog2(float x) {
#if __has_builtin(__builtin_amdgcn_logf)
  return __builtin_amdgcn_logf(x);      // v_log_f32 (base-2) — exactly what we want
#else
  return __log2f(x);
#endif
}

__device__ __forceinline__ void accum(float& acc, float aa, float bb) {
  float s = aa + bb;                               // v_add_f32
  acc = __builtin_fmaf(s, flog2(s), acc);          // v_log_f32 + v_fmac_f32
}

extern "C" __global__ void __launch_bounds__(256)
jsd_kernel(const float* __restrict__ a, const float* __restrict__ b,
           float* __restrict__ out) {
  extern __shared__ __align__(16) float smem[];
  const int tid  = threadIdx.x;
  const int lane = tid & 31;          // wave32
  const int wv   = tid >> 5;          // 8 waves
  const int half = lane >> 4;         // WMMA C layout: lanes 16-31 hold rows M+8
  const int jl   = lane & 15;

  const int tileI = blockIdx.y * TILE;
  const int tileJ = blockIdx.x * TILE;

  float* aS0 = smem;
  float* bS0 = smem + ROWF;
  float* aS1 = smem + BUFF;
  float* bS1 = smem + BUFF + ROWF;
  float* rent = smem + RENT_OFF;      // [0..63]=Σ a log2 a, [64..127]=Σ b log2 b

  const char* aG = (const char*)(a + (size_t)tileI * DTOT);
  const char* bG = (const char*)(b + (size_t)tileJ * DTOT);

  // async-stage one 64x128 chunk of each matrix into LDS (16 x b128 per thread)
  auto issue = [&](int c, int buf) {
    float* aL = buf ? aS1 : aS0;
    float* bL = buf ? bS1 : bS0;
    const size_t gcol = (size_t)c * (DC * 4);
#pragma unroll
    for (int k = 0; k < 8; ++k) {
      int u = tid + 256 * k;                 // 0..2047 16B units
      int row = u >> 5;
      int col = (u & 31) * 16;
      async_b128(aG + (size_t)row * (DTOT * 4) + gcol + col,
                 (char*)(aL + row * STR) + col);
    }
#pragma unroll
    for (int k = 0; k < 8; ++k) {
      int u = tid + 256 * k;
      int row = u >> 5;
      int col = (u & 31) * 16;
      async_b128(bG + (size_t)row * (DTOT * 4) + gcol + col,
                 (char*)(bL + row * STR) + col);
    }
  };

  // wave -> two 16x16 subtiles sharing the same a-rows
  const int i0  = (wv >> 1) * 16;
  const int j0A = (wv & 1) * 32;
  const int j0B = j0A + 16;

  float accA[8], accB[8];
#pragma unroll
  for (int r = 0; r < 8; ++r) { accA[r] = 0.f; accB[r] = 0.f; }
  float ent = 0.f;

  const int aBase = (i0 + 8 * half) * STR;
  const int bArow = (j0A + jl) * STR;
  const int bBrow = (j0B + jl) * STR;

  // row-entropy assignment: 2 threads per row, rows 0-63 = a-tile, 64-127 = b-tile
  const int erow = tid >> 1;
  const int ecol = (tid & 1) * 64;

  issue(0, 0);

  for (int c = 0; c < NCHUNK; ++c) {
    if (c + 1 < NCHUNK) { issue(c + 1, (c + 1) & 1); WAIT_ASYNC(16); }
    else                { WAIT_ASYNC(0); }
    __syncthreads();

    const float* aT = (c & 1) ? aS1 : aS0;
    const float* bT = (c & 1) ? bS1 : bS0;

#pragma unroll 4
    for (int d = 0; d < DC; d += 4) {
      const float4 bx = *(const float4*)(bT + bArow + d);   // 16 distinct banks
      const float4 by = *(const float4*)(bT + bBrow + d);
#pragma unroll
      for (int r = 0; r < 8; ++r) {
        const float4 av = *(const float4*)(aT + aBase + r * STR + d);  // 16-lane broadcast
        accum(accA[r], av.x, bx.x); accum(accA[r], av.y, bx.y);
        accum(accA[r], av.z, bx.z); accum(accA[r], av.w, bx.w);
        accum(accB[r], av.x, by.x); accum(accB[r], av.y, by.y);
        accum(accB[r], av.z, by.z); accum(accB[r], av.w, by.w);
      }
    }

    // per-row entropies from the staged tile (3% extra work, no extra HBM traffic)
    {
      const float* es = (tid < 128) ? (aT + erow * STR + ecol)
                                    : (bT + (erow - 64) * STR + ecol);
#pragma unroll 8
      for (int k = 0; k < 64; ++k) {
        float x = es[k];
        ent = __builtin_fmaf(x, flog2(x), ent);
      }
    }
    __syncthreads();
  }

  ent += __shfl_xor(ent, 1, 32);
  if ((tid & 1) == 0) rent[tid >> 1] = ent;
  __syncthreads();

  const float rav = rent[i0 + jl];
  const float rbA = rent[64 + j0A + jl];
  const float rbB = rent[64 + j0B + jl];

  const int giBase = tileI + i0 + 8 * half;
  const int gjA = tileJ + j0A + jl;
  const int gjB = tileJ + j0B + jl;

#ifdef HAVE_WMMA4
  // out = A x B + C  with A = [0.5*ra_i, 1, 0, 0], B = [[1],[0.5*rb_j],[0],[0]],
  // C = 1 - 0.5*acc.  A layout: lanes 0-15 hold M=lane, K=0/1; lanes 16-31 = K=2/3 (zero).
  v2f Am, BmA, BmB;
  Am.x  = half ? 0.f : 0.5f * rav;  Am.y  = half ? 0.f : 1.0f;
  BmA.x = half ? 0.f : 1.0f;        BmA.y = half ? 0.f : 0.5f * rbA;
  BmB.x = half ? 0.f : 1.0f;        BmB.y = half ? 0.f : 0.5f * rbB;
  v8f cA, cB;
#pragma unroll
  for (int r = 0; r < 8; ++r) {
    cA[r] = __builtin_fmaf(-0.5f, accA[r], 1.0f);
    cB[r] = __builtin_fmaf(-0.5f, accB[r], 1.0f);
  }
  v8f dA = __builtin_amdgcn_wmma_f32_16x16x4_f32(false, Am, false, BmA, (short)0, cA, false, false);
  v8f dB = __builtin_amdgcn_wmma_f32_16x16x4_f32(false, Am, false, BmB, (short)0, cB, false, false);
#pragma unroll
  for (int r = 0; r < 8; ++r) {
    out[(size_t)(giBase + r) * 1024 + gjA] = dA[r];
    out[(size_t)(giBase + r) * 1024 + gjB] = dB[r];
  }
#else
#pragma unroll
  for (int r = 0; r < 8; ++r) {
    float ra_r = rent[i0 + 8 * half + r];
    float oA = __builtin_fmaf(-0.5f, accA[r], 1.0f) + 0.5f * (ra_r + rbA);
    float oB = __builtin_fmaf(-0.5f, accB[r], 1.0f) + 0.5f * (ra_r + rbB);
    out[(size_t)(giBase + r) * 1024 + gjA] = oA;
    out[(size_t)(giBase + r) * 1024 + gjB] = oB;
  }
#endif
  (void)rav;
}

extern "C" void kernel_launch(void* const* d_in, const int* in_sizes, int n_in,
                              void* d_out, int out_size, void* d_ws, size_t ws_size,
                              hipStream_t stream) {
  (void)in_sizes; (void)n_in; (void)out_size; (void)d_ws; (void)ws_size;
  const float* a = (const float*)d_in[0];
  const float* b = (const float*)d_in[1];
  float* out = (float*)d_out;

  // allow >48KB dynamic LDS (CDNA5 WGP: up to 320KB per workgroup)
  (void)hipFuncSetAttribute((const void*)jsd_kernel,
                            hipFuncAttributeMaxDynamicSharedMemorySize, SMEM_BYTES);

  dim3 grid(1024 / TILE, 1024 / TILE);   // 16 x 16 tiles
  jsd_kernel<<<grid, 256, SMEM_BYTES, stream>>>(a, b, out);
}